// CTCAlignDistillLoss_75548474737115
// MI455X (gfx1250) — compile-verified
//
#include <hip/hip_runtime.h>
#include <cstdint>

#define BB 16
#define TT 512
#define VV 8000
#define LSM_C 0.1f
#define WSOFT_C 0.5f

constexpr int NV4  = VV / 4;                   // 2000 float4 per row
constexpr int RBLK = 256;                      // 8 waves (wave32)
constexpr int NCH  = (NV4 + RBLK - 1) / RBLK;  // 8 chunks
constexpr int NBUF = 3;                        // 3-deep async ring

// ---------------- async-to-LDS path (gfx1250) --------------------------------
#if __has_builtin(__builtin_amdgcn_global_load_async_to_lds_b128) && \
    __has_builtin(__builtin_amdgcn_s_wait_asynccnt)
#define USE_ASYNC_LDS 1
#else
#define USE_ASYNC_LDS 0
#endif

#if USE_ASYNC_LDS
typedef int v4i __attribute__((ext_vector_type(4)));
typedef __attribute__((address_space(1))) v4i GV4;
typedef __attribute__((address_space(3))) v4i LV4;
template <int CPOL>
__device__ __forceinline__ void async_cp16(const void* gsrc, void* ldst) {
  GV4* g = (GV4*)(uintptr_t)gsrc;
  LV4* l = (LV4*)(uint32_t)(uintptr_t)ldst;   // low 32 bits = LDS byte offset
  __builtin_amdgcn_global_load_async_to_lds_b128(g, l, 0, CPOL);
}
#endif

// ---------------- kernel 1: run/label metadata -------------------------------
// meta[b,t] = nonblank ? (lm+1) : 0 ; nex[b] = #nonblank frames
__global__ void meta_kernel(const int* __restrict__ aligns,
                            const int* __restrict__ xlens,
                            int* __restrict__ meta, int* __restrict__ nex) {
  const int b = blockIdx.x;
  if (threadIdx.x != 0) return;
  const int xl = xlens[b];
  int run = -1, cnt = 0, prev = 0;
  for (int t = 0; t < TT; ++t) {
    int a  = (t < xl) ? aligns[b * TT + t] : 0;
    int nb = (a != 0);
    if (nb && a != prev) run++;
    prev = a;
    int lm = run < 0 ? 0 : run;
    meta[b * TT + t] = nb ? (lm + 1) : 0;
    cnt += nb;
  }
  nex[b] = cnt;
}

// ---------------- online accumulator update ----------------------------------
__device__ __forceinline__ void upd(float l, float s, float& mx, float& se,
                                    float& sum_l, float& dot, float& sum_s) {
  float nm = fmaxf(mx, l);
  se = se * __expf(mx - nm) + __expf(l - nm);
  mx = nm;
  sum_l += l;
  dot   = fmaf(s, l, dot);
  sum_s += s;
}

// ---------------- kernel 2: per-(b,t) row reduction --------------------------
__global__ __launch_bounds__(RBLK) void row_kernel(
    const float* __restrict__ logits, const int* __restrict__ ys,
    const float* __restrict__ soft, const int* __restrict__ meta,
    float* __restrict__ fs, float* __restrict__ fh) {
  const int row = blockIdx.x;          // b*TT + t
  const int tid = threadIdx.x;
  const int m   = meta[row];
  if (m == 0) {                        // blank frame: contributes nothing
    if (tid == 0) { fs[row] = 0.0f; fh[row] = 0.0f; }
    return;
  }
  const int b  = row / TT;
  const int lm = m - 1;
  const float*  lrow = logits + (size_t)row * VV;
  const float*  srow = soft + ((size_t)b * TT + lm) * VV;
  const float4* l4   = (const float4*)lrow;
  const float4* s4   = (const float4*)srow;

  __builtin_prefetch(lrow, 0, 0);      // global_prefetch_b8
  __builtin_prefetch(srow, 0, 0);

  float mx = -__builtin_inff();
  float se = 0.0f, sum_l = 0.0f, dot = 0.0f, sum_s = 0.0f;

#if USE_ASYNC_LDS
  __shared__ float4 bufL[NBUF][RBLK];
  __shared__ float4 bufS[NBUF][RBLK];
  // logits stream: non-temporal (read-once, 262 MB); soft rows: default RT (reused)
  {  // prime chunks 0 and 1 (indices < 512 <= NV4, always valid)
    async_cp16<1>(&l4[tid], &bufL[0][tid]);
    async_cp16<0>(&s4[tid], &bufS[0][tid]);
    async_cp16<1>(&l4[RBLK + tid], &bufL[1][tid]);
    async_cp16<0>(&s4[RBLK + tid], &bufS[1][tid]);
  }
  for (int c = 0; c < NCH; ++c) {
    if (c + 2 < NCH) {
      int i4n = (c + 2) * RBLK + tid;
      if (i4n >= NV4) i4n = NV4 - 1;   // clamp tail (safe duplicate read)
      async_cp16<1>(&l4[i4n], &bufL[(c + 2) % NBUF][tid]);
      async_cp16<0>(&s4[i4n], &bufS[(c + 2) % NBUF][tid]);
      __builtin_amdgcn_s_wait_asynccnt(4);   // chunk c's 2 loads complete
    } else if (c + 1 < NCH) {
      __builtin_amdgcn_s_wait_asynccnt(2);
    } else {
      __builtin_amdgcn_s_wait_asynccnt(0);
    }
    asm volatile("" ::: "memory");
    const int i4 = c * RBLK + tid;
    if (i4 < NV4) {
      float4 lv = bufL[c % NBUF][tid]; // each lane reads only its own copy
      float4 sv = bufS[c % NBUF][tid];
      upd(lv.x, sv.x, mx, se, sum_l, dot, sum_s);
      upd(lv.y, sv.y, mx, se, sum_l, dot, sum_s);
      upd(lv.z, sv.z, mx, se, sum_l, dot, sum_s);
      upd(lv.w, sv.w, mx, se, sum_l, dot, sum_s);
    }
  }
#else
  for (int i4 = tid; i4 < NV4; i4 += RBLK) {
    int pf = i4 + 2 * RBLK;
    if (pf < NV4) { __builtin_prefetch(&l4[pf], 0, 0); __builtin_prefetch(&s4[pf], 0, 0); }
    float4 lv = l4[i4];
    float4 sv = s4[i4];
    upd(lv.x, sv.x, mx, se, sum_l, dot, sum_s);
    upd(lv.y, sv.y, mx, se, sum_l, dot, sum_s);
    upd(lv.z, sv.z, mx, se, sum_l, dot, sum_s);
    upd(lv.w, sv.w, mx, se, sum_l, dot, sum_s);
  }
#endif

  // wave32 reduction
#pragma unroll
  for (int off = 16; off >= 1; off >>= 1) {
    float om = __shfl_xor(mx, off, 32);
    float os = __shfl_xor(se, off, 32);
    float nm = fmaxf(mx, om);
    se = se * __expf(mx - nm) + os * __expf(om - nm);
    mx = nm;
    sum_l += __shfl_xor(sum_l, off, 32);
    dot   += __shfl_xor(dot,   off, 32);
    sum_s += __shfl_xor(sum_s, off, 32);
  }
  __shared__ float wm[8], wse[8], wsl[8], wd[8], wss[8];
  const int wave = tid >> 5;
  if ((tid & 31) == 0) {
    wm[wave] = mx; wse[wave] = se; wsl[wave] = sum_l; wd[wave] = dot; wss[wave] = sum_s;
  }
  __syncthreads();
  if (tid == 0) {
    float M = wm[0], S = wse[0], SL = wsl[0], D = wd[0], SS = wss[0];
#pragma unroll
    for (int w = 1; w < 8; ++w) {
      float nm = fmaxf(M, wm[w]);
      S = S * __expf(M - nm) + wse[w] * __expf(wm[w] - nm);
      M = nm;
      SL += wsl[w]; D += wd[w]; SS += wss[w];
    }
    const float logZ       = M + __logf(S);
    const int   y          = ys[b * TT + lm];
    const float lpy        = lrow[y] - logZ;
    const float frame_soft = D - logZ * SS;                 // sum_v soft*(logit-logZ)
    const float sum_lp     = SL - (float)VV * logZ;
    const float frame_hard = (1.0f - LSM_C) * lpy +
                             (LSM_C / (float)(VV - 1)) * (sum_lp - lpy);
    fs[row] = frame_soft;
    fh[row] = frame_hard;
  }
}

// ---------------- kernel 3: deterministic final reduction --------------------
__global__ __launch_bounds__(256) void finalize_kernel(
    const float* __restrict__ fs, const float* __restrict__ fh,
    const int* __restrict__ nex, float* __restrict__ out) {
  __shared__ float rs[256], rh[256];
  const int tid = threadIdx.x;
  float acc = 0.0f;
  for (int b = 0; b < BB; ++b) {
    float s = 0.0f, h = 0.0f;
    for (int t = tid; t < TT; t += 256) {
      s += fs[b * TT + t];
      h += fh[b * TT + t];
    }
    rs[tid] = s; rh[tid] = h;
    __syncthreads();
    for (int off = 128; off > 0; off >>= 1) {
      if (tid < off) { rs[tid] += rs[tid + off]; rh[tid] += rh[tid + off]; }
      __syncthreads();
    }
    if (tid == 0) {
      float n = (float)nex[b];
      acc += WSOFT_C * (rs[0] / n) + (1.0f - WSOFT_C) * (rh[0] / n);
    }
    __syncthreads();
  }
  if (tid == 0) out[0] = -acc / (float)BB;
}

// ---------------- launcher ---------------------------------------------------
extern "C" void kernel_launch(void* const* d_in, const int* in_sizes, int n_in,
                              void* d_out, int out_size, void* d_ws, size_t ws_size,
                              hipStream_t stream) {
  (void)in_sizes; (void)n_in; (void)out_size; (void)ws_size;
  const float* logits = (const float*)d_in[0];
  const int*   ys     = (const int*)d_in[1];
  const float* soft   = (const float*)d_in[2];
  const int*   aligns = (const int*)d_in[3];
  const int*   xlens  = (const int*)d_in[4];
  // d_in[5] = ylens: derivable, unused

  int*   meta = (int*)d_ws;
  int*   nex  = meta + BB * TT;
  float* fs   = (float*)(nex + BB);
  float* fh   = fs + BB * TT;

  meta_kernel<<<BB, 32, 0, stream>>>(aligns, xlens, meta, nex);
  row_kernel<<<BB * TT, RBLK, 0, stream>>>(logits, ys, soft, meta, fs, fh);
  finalize_kernel<<<1, 256, 0, stream>>>(fs, fh, nex, (float*)d_out);
}